// SMPL_layer_6124623364114
// MI455X (gfx1250) — compile-verified
//
#include <hip/hip_runtime.h>
#include <hip/hip_bf16.h>

typedef float v2f __attribute__((ext_vector_type(2)));
typedef float v8f __attribute__((ext_vector_type(8)));

#define NV     6890
#define NJ     24
#define NB     10
#define NPB    207
#define BATCH  1024
#define NCOLS  (NV * 3)          // 20670
#define KPAD   208
#define TILES_M (BATCH / 16)          // 64
#define TILES_N4 ((NCOLS + 63) / 64)  // 323 (16x64 tile per wave)

// ---- workspace offsets (in floats) ----
#define WS_JS   0                 // 720
#define WS_J0   720               // 72
#define WS_PF   1024              // 1024*208 = 212992
#define WS_J    214016            // 1024*24*3 = 73728
#define WS_A    287744            // 1024*24*12 = 294912
#define WS_J17  582656            // 1024*17*3 = 52224

// ---- output offsets (in floats) ----
#define O_VERT  0                         // 1024*6890*3
#define O_JOINT 21166080                  // 1024*24*3
#define O_J17   21239808                  // 1024*17*3
#define O_ROT   21292032                  // 1024*24*9

__constant__ int c_parents[NJ] = {0,0,0,0,1,2,3,4,5,6,7,8,9,9,9,12,13,14,16,17,18,19,20,21};

// K0: JS[j,c,l] = sum_v Jreg[j,v]*shapedirs[v,c,l]; J0[j,c] = sum_v Jreg[j,v]*v_template[v,c]
__global__ void k_js(const float* __restrict__ Jreg, const float* __restrict__ vt,
                     const float* __restrict__ sd, float* __restrict__ ws) {
    int t = blockIdx.x * blockDim.x + threadIdx.x;
    if (t >= NJ * 3) return;
    int j = t / 3, c = t % 3;
    float j0 = 0.f;
    float js[NB];
    for (int l = 0; l < NB; ++l) js[l] = 0.f;
    for (int v = 0; v < NV; ++v) {
        float wv = Jreg[j * NV + v];
        j0 += wv * vt[v * 3 + c];
        const float* s = sd + (size_t)(v * 3 + c) * NB;
        for (int l = 0; l < NB; ++l) js[l] += wv * s[l];
    }
    ws[WS_J0 + t] = j0;
    for (int l = 0; l < NB; ++l) ws[WS_JS + t * NB + l] = js[l];
}

// K1: Rodrigues per (batch, joint) -> rot_mats (out), pose_feature (ws, padded), J (ws)
__global__ void k_pose(const float* __restrict__ pose_aa, const float* __restrict__ betas,
                       const float* __restrict__ go, float* __restrict__ ws,
                       float* __restrict__ out) {
    int t = blockIdx.x * blockDim.x + threadIdx.x;
    if (t >= BATCH * NJ) return;
    int b = t / NJ, j = t % NJ;
    float rv0, rv1, rv2;
    if (j == 0) { rv0 = go[b*3]; rv1 = go[b*3+1]; rv2 = go[b*3+2]; }
    else { const float* p = pose_aa + b*69 + (j-1)*3; rv0 = p[0]; rv1 = p[1]; rv2 = p[2]; }
    float e0 = rv0 + 1e-8f, e1 = rv1 + 1e-8f, e2 = rv2 + 1e-8f;
    float angle = sqrtf(e0*e0 + e1*e1 + e2*e2);
    float inv = 1.0f / angle;
    float rx = rv0 * inv, ry = rv1 * inv, rz = rv2 * inv;
    float cs = cosf(angle), sn = sinf(angle);
    float omc = 1.0f - cs;
    float R[9];
    R[0] = cs + omc*rx*rx;      R[1] = omc*rx*ry - sn*rz;  R[2] = omc*rx*rz + sn*ry;
    R[3] = omc*ry*rx + sn*rz;   R[4] = cs + omc*ry*ry;     R[5] = omc*ry*rz - sn*rx;
    R[6] = omc*rz*rx - sn*ry;   R[7] = omc*rz*ry + sn*rx;  R[8] = cs + omc*rz*rz;
    float* rout = out + O_ROT + (size_t)(b*NJ + j) * 9;
    for (int i = 0; i < 9; ++i) rout[i] = R[i];
    if (j > 0) {
        float* q = ws + WS_PF + (size_t)b * KPAD + (j-1) * 9;
        for (int i = 0; i < 9; ++i) q[i] = R[i] - (((i & 3) == 0) ? 1.0f : 0.0f);
    } else {
        ws[WS_PF + (size_t)b * KPAD + 207] = 0.0f;   // K padding
    }
    // J[b,j,c] = J0[j,c] + sum_l betas[b,l]*JS[j,c,l]
    const float* be = betas + b * NB;
    for (int c = 0; c < 3; ++c) {
        float acc = ws[WS_J0 + j*3 + c];
        const float* js = ws + WS_JS + (j*3 + c) * NB;
        for (int l = 0; l < NB; ++l) acc += be[l] * js[l];
        ws[WS_J + (size_t)(b*NJ + j)*3 + c] = acc;
    }
}

// K2: kinematic chain per batch -> A (ws), joints (out, root-relative)
__global__ void k_chain(float* __restrict__ ws, float* __restrict__ out) {
    int b = blockIdx.x * blockDim.x + threadIdx.x;
    if (b >= BATCH) return;
    float G[NJ][12];   // global 3x4 transforms
    const float* rot = out + O_ROT + (size_t)b * NJ * 9;
    const float* Jb  = ws + WS_J + (size_t)b * NJ * 3;
    for (int j = 0; j < NJ; ++j) {
        const float* R = rot + j * 9;
        float t0, t1, t2;
        if (j == 0) { t0 = Jb[0]; t1 = Jb[1]; t2 = Jb[2]; }
        else {
            int p = c_parents[j];
            t0 = Jb[j*3+0] - Jb[p*3+0];
            t1 = Jb[j*3+1] - Jb[p*3+1];
            t2 = Jb[j*3+2] - Jb[p*3+2];
        }
        if (j == 0) {
            for (int r = 0; r < 3; ++r) {
                G[0][4*r+0] = R[3*r+0]; G[0][4*r+1] = R[3*r+1]; G[0][4*r+2] = R[3*r+2];
            }
            G[0][3] = t0; G[0][7] = t1; G[0][11] = t2;
        } else {
            int p = c_parents[j];
            for (int r = 0; r < 3; ++r) {
                float p0 = G[p][4*r+0], p1 = G[p][4*r+1], p2 = G[p][4*r+2], p3 = G[p][4*r+3];
                G[j][4*r+0] = p0*R[0] + p1*R[3] + p2*R[6];
                G[j][4*r+1] = p0*R[1] + p1*R[4] + p2*R[7];
                G[j][4*r+2] = p0*R[2] + p1*R[5] + p2*R[8];
                G[j][4*r+3] = p0*t0 + p1*t1 + p2*t2 + p3;
            }
        }
    }
    // joints output (root-relative) + A = [Rglob | t - Rglob @ J]
    float r0 = G[0][3], r1 = G[0][7], r2 = G[0][11];
    for (int j = 0; j < NJ; ++j) {
        float* jo = out + O_JOINT + (size_t)(b*NJ + j) * 3;
        jo[0] = G[j][3] - r0; jo[1] = G[j][7] - r1; jo[2] = G[j][11] - r2;
        float jx = Jb[j*3+0], jy = Jb[j*3+1], jz = Jb[j*3+2];
        float* Aj = ws + WS_A + (size_t)(b*NJ + j) * 12;
        for (int r = 0; r < 3; ++r) {
            Aj[4*r+0] = G[j][4*r+0]; Aj[4*r+1] = G[j][4*r+1]; Aj[4*r+2] = G[j][4*r+2];
            Aj[4*r+3] = G[j][4*r+3] - (G[j][4*r+0]*jx + G[j][4*r+1]*jy + G[j][4*r+2]*jz);
        }
    }
}

// K3: WMMA f32 GEMM: v_posed = pose_feature @ posedirs + v_template + betas@shapedirs
// M=1024(batch) x N=20670 x K=207 ; 16x64 tile per wave (4 accumulators), K step 4.
// The 8 waves of a block share one tn (identical B addresses -> WGP$ hits) and take
// 8 consecutive tm values. Branch-free main loop (even trip count 50) + 2 peeled
// K-tail steps; load columns clamped in-bounds; stores guarded in epilogue.
__global__ __launch_bounds__(256) void k_gemm(const float* __restrict__ ws,
                                              const float* __restrict__ pd,
                                              const float* __restrict__ vt,
                                              const float* __restrict__ sd,
                                              const float* __restrict__ betas,
                                              float* __restrict__ out) {
    int lane = threadIdx.x & 31;
    int wave = threadIdx.x >> 5;
    int tn  = blockIdx.x % TILES_N4;           // shared by all 8 waves of the block
    int tm  = (blockIdx.x / TILES_N4) * 8 + wave;
    if (tm >= TILES_M) return;
    int ml = lane & 15;
    int hi = lane >> 4;
    int row = tm * 16 + ml;              // batch index (A row)
    int col0 = tn * 64 + ml;             // first of 4 output columns (B cols)
    int cols[4];
    int colc[4];
    bool cok[4];
    #pragma unroll
    for (int t = 0; t < 4; ++t) {
        cols[t] = col0 + 16 * t;
        cok[t]  = (cols[t] < NCOLS);
        colc[t] = cok[t] ? cols[t] : (NCOLS - 1);   // clamp: loads always in-bounds
    }
    const float* arow = ws + WS_PF + (size_t)row * KPAD;
    v8f acc[4] = {};
    for (int kb = 0; kb < 50; ++kb) {    // even trip count; k <= 199 < 207
        int k0 = kb * 4 + hi * 2;
        v2f a;
        a.x = arow[k0];
        a.y = arow[k0 + 1];
        const float* prow0 = pd + (size_t)k0 * NCOLS;
        const float* prow1 = prow0 + NCOLS;
        #pragma unroll
        for (int t = 0; t < 4; ++t) {
            v2f bb;
            bb.x = prow0[colc[t]];
            bb.y = prow1[colc[t]];
            acc[t] = __builtin_amdgcn_wmma_f32_16x16x4_f32(false, a, false, bb,
                                                           (short)0, acc[t], false, false);
        }
    }
    {   // peeled K step: kb = 50, k = 200..203, all valid
        int k0 = 200 + hi * 2;
        v2f a;
        a.x = arow[k0];
        a.y = arow[k0 + 1];
        const float* prow0 = pd + (size_t)k0 * NCOLS;
        const float* prow1 = prow0 + NCOLS;
        #pragma unroll
        for (int t = 0; t < 4; ++t) {
            v2f bb;
            bb.x = prow0[colc[t]];
            bb.y = prow1[colc[t]];
            acc[t] = __builtin_amdgcn_wmma_f32_16x16x4_f32(false, a, false, bb,
                                                           (short)0, acc[t], false, false);
        }
    }
    {   // peeled K tail: kb = 51, k0 = 204 + 2*hi; only k = 207 is invalid
        int k0 = 204 + hi * 2;
        v2f a;
        a.x = arow[k0];
        a.y = arow[k0 + 1];              // PF padded to 208 with zeros: in-bounds
        float mask1 = ((k0 + 1) < NPB) ? 1.0f : 0.0f;
        int k1c = ((k0 + 1) < NPB) ? (k0 + 1) : (NPB - 1);   // clamped row
        const float* prow0 = pd + (size_t)k0 * NCOLS;
        const float* prow1 = pd + (size_t)k1c * NCOLS;
        #pragma unroll
        for (int t = 0; t < 4; ++t) {
            v2f bb;
            bb.x = prow0[colc[t]];
            bb.y = prow1[colc[t]] * mask1;
            acc[t] = __builtin_amdgcn_wmma_f32_16x16x4_f32(false, a, false, bb,
                                                           (short)0, acc[t], false, false);
        }
    }
    // epilogue: fuse v_template + betas@shapedirs (shapedirs index is exactly col*10+l)
    float sdv[4][NB];
    float vtv[4];
    #pragma unroll
    for (int t = 0; t < 4; ++t) {
        vtv[t] = vt[colc[t]];
        for (int l = 0; l < NB; ++l) sdv[t][l] = sd[(size_t)colc[t] * NB + l];
    }
    #pragma unroll
    for (int r = 0; r < 8; ++r) {
        int b = tm * 16 + r + hi * 8;
        float be[NB];
        const float* bep = betas + (size_t)b * NB;
        for (int l = 0; l < NB; ++l) be[l] = bep[l];
        #pragma unroll
        for (int t = 0; t < 4; ++t) {
            if (cok[t]) {
                float val = acc[t][r] + vtv[t];
                for (int l = 0; l < NB; ++l) val += be[l] * sdv[t][l];
                out[O_VERT + (size_t)b * NCOLS + cols[t]] = val;
            }
        }
    }
}

// K4: linear blend skinning, in-place on vertices region (elementwise per (b,v))
__global__ void k_skin(const float* __restrict__ w, const float* __restrict__ ws,
                       float* __restrict__ out) {
    int t = blockIdx.x * blockDim.x + threadIdx.x;
    if (t >= BATCH * NV) return;
    int b = t / NV, v = t % NV;
    float T[12];
    for (int i = 0; i < 12; ++i) T[i] = 0.0f;
    const float* wb = w + (size_t)v * NJ;
    const float* Ab = ws + WS_A + (size_t)b * NJ * 12;
    for (int j = 0; j < NJ; ++j) {
        float wj = wb[j];
        const float* Aj = Ab + j * 12;
        for (int i = 0; i < 12; ++i) T[i] += wj * Aj[i];
    }
    float* p = out + O_VERT + (size_t)b * NCOLS + (size_t)v * 3;
    float x = p[0], y = p[1], z = p[2];
    p[0] = T[0]*x + T[1]*y + T[2]*z  + T[3];
    p[1] = T[4]*x + T[5]*y + T[6]*z  + T[7];
    p[2] = T[8]*x + T[9]*y + T[10]*z + T[11];
}

// K5: j17 = J_regressor_h36m @ verts
__global__ void k_j17(const float* __restrict__ Jr17, const float* __restrict__ out,
                      float* __restrict__ ws) {
    int t = blockIdx.x * blockDim.x + threadIdx.x;
    if (t >= BATCH * 51) return;
    int b = t / 51, q = t % 51;
    int j = q / 3, c = q % 3;
    const float* jr = Jr17 + (size_t)j * NV;
    const float* vb = out + O_VERT + (size_t)b * NCOLS + c;
    float acc = 0.0f;
    for (int v = 0; v < NV; ++v) acc += jr[v] * vb[(size_t)v * 3];
    ws[WS_J17 + t] = acc;
}

// K6: vertices -= root17 (in place)
__global__ void k_fin_v(const float* __restrict__ ws, float* __restrict__ out) {
    size_t t = (size_t)blockIdx.x * blockDim.x + threadIdx.x;
    if (t >= (size_t)BATCH * NCOLS) return;
    int b = (int)(t / NCOLS);
    int c = (int)(t % 3);
    out[O_VERT + t] -= ws[WS_J17 + (size_t)b * 51 + c];
}

// K7: joints_from_verts output = j17 - root17
__global__ void k_fin_j(const float* __restrict__ ws, float* __restrict__ out) {
    int t = blockIdx.x * blockDim.x + threadIdx.x;
    if (t >= BATCH * 51) return;
    int b = t / 51;
    int c = (t % 51) % 3;
    out[O_J17 + t] = ws[WS_J17 + t] - ws[WS_J17 + b * 51 + c];
}

extern "C" void kernel_launch(void* const* d_in, const int* in_sizes, int n_in,
                              void* d_out, int out_size, void* d_ws, size_t ws_size,
                              hipStream_t stream) {
    const float* pose_aa = (const float*)d_in[0];
    const float* betas   = (const float*)d_in[1];
    const float* go      = (const float*)d_in[2];
    const float* vt      = (const float*)d_in[3];
    const float* sd      = (const float*)d_in[4];
    const float* pd      = (const float*)d_in[5];
    const float* Jreg    = (const float*)d_in[6];
    const float* Jr17    = (const float*)d_in[7];
    const float* lbsw    = (const float*)d_in[8];
    float* out = (float*)d_out;
    float* ws  = (float*)d_ws;

    k_js<<<1, 128, 0, stream>>>(Jreg, vt, sd, ws);
    k_pose<<<(BATCH * NJ + 255) / 256, 256, 0, stream>>>(pose_aa, betas, go, ws, out);
    k_chain<<<(BATCH + 255) / 256, 256, 0, stream>>>(ws, out);
    {
        int nblocks = (TILES_M / 8) * TILES_N4;   // 8 * 323 = 2584
        k_gemm<<<nblocks, 256, 0, stream>>>(ws, pd, vt, sd, betas, out);
    }
    k_skin<<<(BATCH * NV + 255) / 256, 256, 0, stream>>>(lbsw, ws, out);
    k_j17<<<(BATCH * 51 + 255) / 256, 256, 0, stream>>>(Jr17, out, ws);
    {
        size_t n = (size_t)BATCH * NCOLS;
        k_fin_v<<<(unsigned)((n + 255) / 256), 256, 0, stream>>>(ws, out);
    }
    k_fin_j<<<(BATCH * 51 + 255) / 256, 256, 0, stream>>>(ws, out);
}